// PeriDynamicAttention_43980465111629
// MI455X (gfx1250) — compile-verified
//
#include <hip/hip_runtime.h>
#include <hip/hip_bf16.h>

// ---------------------------------------------------------------------------
// PeriDynamicAttention for MI455X (gfx1250, wave32, WMMA + async-to-LDS)
//
//   B=2, T=2048, C=1024, NH=16, HS=64, BD=16, HORIZON=64
//
// fused = (disp[s]-disp[t]) @ Wf^T + b is linear in disp, so with
// Wcomb = Wf @ W_disp_h we have P = x @ Wcomb^T and fused[t,w] = P[s]-P[t]+b.
// The 268MB strain tensor and windowed batched GEMM disappear.
// All large GEMMs run on V_WMMA_F32_16X16X4_F32 with LDS tiles staged by
// GLOBAL_LOAD_ASYNC_TO_LDS_B128 (ASYNCcnt) in a double-buffered pipeline.
// ---------------------------------------------------------------------------

typedef __attribute__((ext_vector_type(2))) float v2f;
typedef __attribute__((ext_vector_type(8))) float v8f;
typedef int v4i_ __attribute__((vector_size(16)));   // matches builtin param type

#define B_   2
#define T_   2048
#define C_   1024
#define NH_  16
#define HS_  64
#define BD_  16
#define WIN  64
#define M_   (B_ * T_)          // 4096 rows for the big GEMMs
#define PC_  (NH_ * 2 * BD_)    // 512 = P row length

#define KC   32                 // K-chunk staged in LDS
#define LPAD 36                 // LDS row stride (floats): 144B -> 16B aligned,
                                // 36*r mod 64 distinct for r=0..15 (no bank conflicts)

#if __has_builtin(__builtin_amdgcn_global_load_async_to_lds_b128) && \
    __has_builtin(__builtin_amdgcn_s_wait_asynccnt)
#define HAVE_ASYNC_LDS 1
// global (as1) / LDS (as3) pointer casts with the exact pointee type the builtin wants
#define GAS_PTR(p) ((__attribute__((address_space(1))) v4i_*)(v4i_*)(p))
#define LAS_PTR(p) ((__attribute__((address_space(3))) v4i_*)(v4i_*)(p))
#else
#define HAVE_ASYNC_LDS 0
#endif

__device__ __forceinline__ float gelu_exact(float x) {
    return 0.5f * x * (1.0f + erff(x * 0.70710678118654752440f));
}

// ---------------------------------------------------------------------------
// Generic Y = X @ Wt^T GEMM on the WMMA pipe.
//   X: (M,K) row-major, Wt: (N,K) row-major, Y: (M,N) row-major.
//   Requires M%64==0, N%64==0, K%KC==0 (all satisfied here).
//   Block = 256 threads = 8 waves; block tile 64x64; wave tile 16x32.
//   Double-buffered K-chunks, staged with async global->LDS B128 copies.
// ---------------------------------------------------------------------------
__global__ __launch_bounds__(256) void gemm_xwt_wmma(
    const float* __restrict__ X, const float* __restrict__ Wt,
    float* __restrict__ Y, int M, int N, int K)
{
    __shared__ __attribute__((aligned(16))) float Xs[2][64][LPAD];
    __shared__ __attribute__((aligned(16))) float Ws[2][64][LPAD];

    const int tid  = threadIdx.x;
    const int lane = tid & 31;
    const int wave = tid >> 5;
    const int bm   = blockIdx.y * 64;
    const int bn   = blockIdx.x * 64;
    const int wm   = (wave >> 1) * 16;   // 0,16,32,48
    const int wn   = (wave & 1) * 32;    // 0,32

    // cooperative staging map: 64 rows x 32 cols, 8 floats (2 x b128) per thread
    const int lrow = tid >> 2;           // 0..63
    const int lcol = (tid & 3) * 8;      // 0,8,16,24

    v8f c0 = {};
    v8f c1 = {};

    // stage one 64x32 X-tile and 64x32 W-tile into buffer `buf`
    auto stage_tile = [&](int buf, int kbase) {
        const float* xg = X  + (size_t)(bm + lrow) * K + kbase + lcol;
        const float* wg = Wt + (size_t)(bn + lrow) * K + kbase + lcol;
        float* xl = &Xs[buf][lrow][lcol];
        float* wl = &Ws[buf][lrow][lcol];
#if HAVE_ASYNC_LDS
        // 4 wave-level async instructions per stage (tracked by ASYNCcnt)
        __builtin_amdgcn_global_load_async_to_lds_b128(GAS_PTR(xg),     LAS_PTR(xl),     0, 0);
        __builtin_amdgcn_global_load_async_to_lds_b128(GAS_PTR(xg + 4), LAS_PTR(xl + 4), 0, 0);
        __builtin_amdgcn_global_load_async_to_lds_b128(GAS_PTR(wg),     LAS_PTR(wl),     0, 0);
        __builtin_amdgcn_global_load_async_to_lds_b128(GAS_PTR(wg + 4), LAS_PTR(wl + 4), 0, 0);
#else
#pragma unroll
        for (int i = 0; i < 8; ++i) xl[i] = xg[i];
#pragma unroll
        for (int i = 0; i < 8; ++i) wl[i] = wg[i];
#endif
    };

    int cur = 0;
    stage_tile(0, 0);

    for (int k0 = 0; k0 < K; k0 += KC) {
        const bool has_next = (k0 + KC) < K;
        if (has_next) stage_tile(cur ^ 1, k0 + KC);   // prefetch next chunk (async)
#if HAVE_ASYNC_LDS
        // drain the older async group (4 ops), keep the 4 just issued in flight
        if (has_next) __builtin_amdgcn_s_wait_asynccnt(4);
        else          __builtin_amdgcn_s_wait_asynccnt(0);
#endif
        __syncthreads();   // tile[cur] visible to all waves

#pragma unroll
        for (int kk = 0; kk < KC; kk += 4) {
            // A 16x4 f32 fragment: lanes 0-15 hold K=kk,kk+1; lanes 16-31 K=kk+2,kk+3
            const int kr  = kk + ((lane < 16) ? 0 : 2);
            const int mr  = wm + (lane & 15);
            v2f a, b0, b1;
            a.x = Xs[cur][mr][kr];
            a.y = Xs[cur][mr][kr + 1];
            // B 4x16 fragment (B[k][n] = Wt[n][k]), same half-wave K split
            const int nr0 = wn + (lane & 15);
            b0.x = Ws[cur][nr0][kr];
            b0.y = Ws[cur][nr0][kr + 1];
            const int nr1 = nr0 + 16;
            b1.x = Ws[cur][nr1][kr];
            b1.y = Ws[cur][nr1][kr + 1];
            c0 = __builtin_amdgcn_wmma_f32_16x16x4_f32(false, a, false, b0,
                                                       (short)0, c0, false, false);
            c1 = __builtin_amdgcn_wmma_f32_16x16x4_f32(false, a, false, b1,
                                                       (short)0, c1, false, false);
        }
        __syncthreads();   // all waves done reading tile[cur]; safe to overwrite
        cur ^= 1;
    }

    // C/D layout: VGPR r <-> rows r (lanes 0-15) and r+8 (lanes 16-31), col = lane%16
    const int row = bm + wm + ((lane < 16) ? 0 : 8);
    const int col = bn + wn + (lane & 15);
#pragma unroll
    for (int r = 0; r < 8; ++r) {
        Y[(size_t)(row + r) * N + col]      = c0[r];
        Y[(size_t)(row + r) * N + col + 16] = c1[r];
    }
}

// ---------------------------------------------------------------------------
// Wcomb[h*32+j, c] = sum_k Wf[j,k] * W_disp[h*16+k, c]   (512 x 1024)
// ---------------------------------------------------------------------------
__global__ __launch_bounds__(256) void build_wcomb(
    const float* __restrict__ Wdisp,   // (256,1024)
    const float* __restrict__ Wf,      // (32,16)
    float* __restrict__ Wcomb)         // (512,1024)
{
    const int gid = blockIdx.x * 256 + threadIdx.x;   // 512*1024 threads
    const int c   = gid & (C_ - 1);
    const int row = gid >> 10;
    const int h   = row >> 5;
    const int j   = row & 31;
    float acc = 0.0f;
#pragma unroll
    for (int k = 0; k < BD_; ++k)
        acc = fmaf(Wf[j * BD_ + k], Wdisp[(h * BD_ + k) * C_ + c], acc);
    Wcomb[gid] = acc;
}

// ---------------------------------------------------------------------------
// pos_feat[w, j] = sum_k rel[w,k] * ppw[j,k]   (64 x 16)
// ---------------------------------------------------------------------------
__global__ __launch_bounds__(256) void build_posf(
    const float* __restrict__ rel,     // (64,16)
    const float* __restrict__ ppw,     // (16,16)
    float* __restrict__ posf)          // (64,16)
{
    const int gid = blockIdx.x * 256 + threadIdx.x;   // 1024 threads
    const int w = gid >> 4;
    const int j = gid & 15;
    float acc = 0.0f;
#pragma unroll
    for (int k = 0; k < BD_; ++k)
        acc = fmaf(rel[w * BD_ + k], ppw[j * BD_ + k], acc);
    posf[gid] = acc;
}

// ---------------------------------------------------------------------------
// Window attention: one wave per (b,h,t).
//   fused[t,w]   = P[s] - P[t] + bias,  s = t-63+w, valid iff s >= 0
//   bond_logit   = gelu(fused[:16] + pos_feat[w]) . bond_w
//   damage       = sigmoid(gelu(fused[16:]) . dmg_w + dmg_b)
//   softmax over valid w, then attn_out[b,t,h,:] = sum_w wgt * val[b,s,h,:]
// ---------------------------------------------------------------------------
__global__ __launch_bounds__(256) void attn_window(
    const float* __restrict__ P,       // (B*T, 512) = (b,t,h,32)
    const float* __restrict__ val,     // (B*T, 1024) = (b,t,h,64)
    const float* __restrict__ posf,    // (64,16)
    const float* __restrict__ fb,      // (32)
    const float* __restrict__ bw,      // (16)
    const float* __restrict__ dw,      // (16)
    const float* __restrict__ db,      // (1)
    float* __restrict__ attn_out)      // (B*T, 1024)
{
    __shared__ float s_posf[WIN * BD_];
    __shared__ float s_fb[2 * BD_];
    __shared__ float s_bw[BD_];
    __shared__ float s_dw[BD_];
    __shared__ float s_pt[8][2 * BD_];
    __shared__ float s_wt[8][WIN];

    const int tid  = threadIdx.x;
    const int lane = tid & 31;
    const int wave = tid >> 5;

    for (int i = tid; i < WIN * BD_; i += 256) s_posf[i] = posf[i];
    if (tid < 2 * BD_) s_fb[tid] = fb[tid];
    if (tid < BD_) { s_bw[tid] = bw[tid]; s_dw[tid] = dw[tid]; }

    // gw = (b*NH + h)*T + t ; consecutive waves -> consecutive t (locality)
    const int gw = blockIdx.x * 8 + wave;
    const int t  = gw & (T_ - 1);
    const int h  = (gw >> 11) & (NH_ - 1);
    const int b  = gw >> 15;
    const float dbias = db[0];

    const size_t prow = (size_t)(b * T_ + t) * PC_ + h * 2 * BD_;
    s_pt[wave][lane] = P[prow + lane];    // 32 lanes load 32 floats of P[t]
    __syncthreads();

    const float NEG_INF = -__builtin_inff();
    float l0 = NEG_INF, l1 = NEG_INF;
#pragma unroll
    for (int half = 0; half < 2; ++half) {
        const int w = lane + half * 32;
        const int s = t - (WIN - 1) + w;
        if (s >= 0) {
            const float* Ps = P + (size_t)(b * T_ + s) * PC_ + h * 2 * BD_;
            float bond = 0.0f, dmg = 0.0f;
#pragma unroll 4
            for (int j = 0; j < BD_; ++j) {
                const float qb = Ps[j]       - s_pt[wave][j]       + s_fb[j]
                                 + s_posf[w * BD_ + j];
                const float qd = Ps[BD_ + j] - s_pt[wave][BD_ + j] + s_fb[BD_ + j];
                bond = fmaf(gelu_exact(qb), s_bw[j], bond);
                dmg  = fmaf(gelu_exact(qd), s_dw[j], dmg);
            }
            const float damage = 1.0f / (1.0f + __expf(-(dmg + dbias)));
            const float lg = bond - 10.0f * damage;
            if (half == 0) l0 = lg; else l1 = lg;
        }
    }

    // wave32 softmax over 64 window slots (2 per lane)
    float m = fmaxf(l0, l1);
#pragma unroll
    for (int off = 16; off > 0; off >>= 1) m = fmaxf(m, __shfl_xor(m, off, 32));
    const float e0 = (l0 == NEG_INF) ? 0.0f : __expf(l0 - m);
    const float e1 = (l1 == NEG_INF) ? 0.0f : __expf(l1 - m);
    float ssum = e0 + e1;
#pragma unroll
    for (int off = 16; off > 0; off >>= 1) ssum += __shfl_xor(ssum, off, 32);
    const float inv = 1.0f / ssum;
    s_wt[wave][lane]      = e0 * inv;
    s_wt[wave][lane + 32] = e1 * inv;
    __syncthreads();

    // weighted sum over val rows; lanes split the 64-wide head dim (2 each)
    float acc0 = 0.0f, acc1 = 0.0f;
    const int w0 = (t >= WIN - 1) ? 0 : (WIN - 1 - t);
    for (int w = w0; w < WIN; ++w) {
        const int s = t - (WIN - 1) + w;
        const float wt = s_wt[wave][w];
        const float* vrow = val + (size_t)(b * T_ + s) * C_ + h * HS_;
        acc0 = fmaf(wt, vrow[lane],      acc0);
        acc1 = fmaf(wt, vrow[lane + 32], acc1);
    }
    float* orow = attn_out + (size_t)(b * T_ + t) * C_ + h * HS_;
    orow[lane]      = acc0;
    orow[lane + 32] = acc1;
}

// ---------------------------------------------------------------------------
// Workspace layout (floats):
//   Wcomb : 512*1024              =  524288
//   posf  : 64*16                 =    1024
//   P     : 4096*512              = 2097152
//   val   : 4096*1024             = 4194304
//   attn  : 4096*1024             = 4194304
// total ~ 42 MB
// ---------------------------------------------------------------------------
#define OFF_WCOMB 0
#define OFF_POSF  524288
#define OFF_P     525312
#define OFF_VAL   2622464
#define OFF_ATTN  6816768

extern "C" void kernel_launch(void* const* d_in, const int* in_sizes, int n_in,
                              void* d_out, int out_size, void* d_ws, size_t ws_size,
                              hipStream_t stream) {
    const float* x     = (const float*)d_in[0];   // (B,T,C)
    const float* Wdisp = (const float*)d_in[1];   // (256,1024)
    const float* Wval  = (const float*)d_in[2];   // (1024,1024)
    const float* rel   = (const float*)d_in[3];   // (64,16)
    const float* Wf    = (const float*)d_in[4];   // (32,16)
    const float* fb    = (const float*)d_in[5];   // (32)
    const float* ppw   = (const float*)d_in[6];   // (16,16)
    const float* bw    = (const float*)d_in[7];   // (1,16)
    const float* dw    = (const float*)d_in[8];   // (1,16)
    const float* db    = (const float*)d_in[9];   // (1)
    const float* Wc    = (const float*)d_in[10];  // (1024,1024)
    float* out = (float*)d_out;

    float* ws    = (float*)d_ws;
    float* Wcomb = ws + OFF_WCOMB;
    float* posf  = ws + OFF_POSF;
    float* P     = ws + OFF_P;
    float* val   = ws + OFF_VAL;
    float* attn  = ws + OFF_ATTN;

    build_wcomb<<<(512 * 1024) / 256, 256, 0, stream>>>(Wdisp, Wf, Wcomb);
    build_posf<<<1024 / 256, 256, 0, stream>>>(rel, ppw, posf);

    // P = x @ Wcomb^T  (4096 x 512 x 1024)
    gemm_xwt_wmma<<<dim3(PC_ / 64, M_ / 64), 256, 0, stream>>>(x, Wcomb, P, M_, PC_, C_);
    // val = x @ Wval^T (4096 x 1024 x 1024)
    gemm_xwt_wmma<<<dim3(C_ / 64, M_ / 64), 256, 0, stream>>>(x, Wval, val, M_, C_, C_);

    // window attention: 65536 waves, 8 per block
    attn_window<<<(B_ * NH_ * T_) / 8, 256, 0, stream>>>(P, val, posf, fb, bw, dw, db, attn);

    // out = attn @ c_proj^T (4096 x 1024 x 1024)
    gemm_xwt_wmma<<<dim3(C_ / 64, M_ / 64), 256, 0, stream>>>(attn, Wc, out, M_, C_, C_);
}